// Gng_62122406969537
// MI455X (gfx1250) — compile-verified
//
#include <hip/hip_runtime.h>

// GNG hyperparameters (match reference)
#define E_B    0.05f
#define E_N    0.006f
#define A_MAXE 50
#define L_INS  100
#define D_FACT 0.995f
#define MAXN   512
#define DIM    784
#define BATCH  256
#define BS     1024   // 32 wave32s, one WGP
#define NW     32

typedef __attribute__((ext_vector_type(2))) float v2f;
typedef __attribute__((ext_vector_type(8))) float v8f;

__device__ __forceinline__ bool lt2(float va, int ia, float vb, int ib) {
  // "a before b" for ascending sort with index tie-break (matches top_k stability)
  return (va < vb) || (va == vb && ia < ib);
}
__device__ __forceinline__ bool gt2(float va, int ia, float vb, int ib) {
  // "a wins argmax over b": strictly greater, ties -> lower index (jnp.argmax first-max)
  return (va > vb) || (va == vb && ia < ib);
}

__global__ __launch_bounds__(BS, 1) void gng_persistent(
    float* __restrict__ nodes,          // [MAXN, DIM]  (d_out, working buffer)
    int* __restrict__ edges,            // [MAXN, MAXN] (workspace)
    const float* __restrict__ images,   // [BATCH, DIM]
    const float* __restrict__ err_init, // [MAXN]
    const int* __restrict__ n_act_init) // [1]
{
  __shared__ float s_img[DIM];
  __shared__ float s_diff[DIM];
  __shared__ float s_d2[MAXN];
  __shared__ float s_err[MAXN];
  __shared__ int   s_nb[MAXN];
  __shared__ float s_red[NW];
  __shared__ float s_xn;
  __shared__ int   s_bmu, s_sec, s_nact, s_q, s_f;

  const int tid  = threadIdx.x;
  const int lane = tid & 31;
  const int wv   = tid >> 5;
  const float INF = __builtin_inff();

  // --- init internal state (err lives entirely in LDS; not an output) ---
  for (int i = tid; i < MAXN; i += BS) s_err[i] = err_init[i];
  if (tid == 0) s_nact = *n_act_init;
  __syncthreads();

  for (int t = 0; t < BATCH; ++t) {
    // ---------- Phase 1: image -> LDS, ||x||^2 ----------
    const float* img = images + (size_t)t * DIM;
    float v = 0.f;
    if (tid < DIM) { float x = img[tid]; s_img[tid] = x; v = x * x; }
    for (int off = 16; off; off >>= 1) v += __shfl_xor(v, off, 32);
    if (lane == 0) s_red[wv] = v;
    __syncthreads();
    if (wv == 0) {
      float s = s_red[lane];
      for (int off = 16; off; off >>= 1) s += __shfl_xor(s, off, 32);
      if (lane == 0) s_xn = s;
    }
    __syncthreads();

    // ---------- Phase 2: distances via f32 WMMA (one 16-node tile / wave) ----------
    {
      const int m0   = wv * 16;
      const int mrow = m0 + (lane & 15);
      const int koff = (lane >> 4) << 1;          // 0 for lanes 0-15, 2 for lanes 16-31
      const float* nrow = nodes + (size_t)mrow * DIM;
      v8f cd = {};   // dot(n, x) tile (B = image replicated across 16 cols)
      v8f cg = {};   // Gram tile N*N^T (A-layout == its own B^T layout for 16x4 f32)
      for (int k0 = 0; k0 < DIM; k0 += 4) {
        const int kk = k0 + koff;
        v2f a; a.x = nrow[kk];   a.y = nrow[kk + 1];
        v2f b; b.x = s_img[kk];  b.y = s_img[kk + 1];
        cd = __builtin_amdgcn_wmma_f32_16x16x4_f32(false, a, false, b, (short)0, cd, false, false);
        cg = __builtin_amdgcn_wmma_f32_16x16x4_f32(false, a, false, a, (short)0, cg, false, false);
      }
      // Diagonal extraction: C layout -> VGPR e, lanes 0-15: (M=e,N=lane); lanes 16-31: (M=e+8,N=lane-16)
      float nrm = 0.f, dt = 0.f; int ml = -1;
#pragma unroll
      for (int e = 0; e < 8; ++e) {
        if (lane == e)      { nrm = cg[e]; dt = cd[e]; ml = e; }
        if (lane == e + 24) { nrm = cg[e]; dt = cd[e]; ml = e + 8; }
      }
      const int na = s_nact;
      if (ml >= 0) {
        const int m = m0 + ml;
        const float d2 = nrm - 2.f * dt + s_xn;
        s_d2[m] = (m < na) ? d2 : INF;
      }
    }
    __syncthreads();

    // ---------- Phase 3: top-2 (wave 0) ----------
    if (wv == 0) {
      float bv = INF, sv = INF; int bi = 0x7fffffff, si = 0x7fffffff;
      for (int i = lane; i < MAXN; i += 32) {
        const float d = s_d2[i];
        if (lt2(d, i, bv, bi))      { sv = bv; si = bi; bv = d; bi = i; }
        else if (lt2(d, i, sv, si)) { sv = d;  si = i; }
      }
      for (int off = 16; off; off >>= 1) {
        const float obv = __shfl_xor(bv, off, 32); const int obi = __shfl_xor(bi, off, 32);
        const float osv = __shfl_xor(sv, off, 32); const int osi = __shfl_xor(si, off, 32);
        if (lt2(obv, obi, bv, bi)) {
          if (lt2(bv, bi, osv, osi)) { sv = bv;  si = bi;  } else { sv = osv; si = osi; }
          bv = obv; bi = obi;
        } else if (lt2(obv, obi, sv, si)) { sv = obv; si = obi; }
      }
      if (lane == 0) { s_bmu = bi; s_sec = si; }
    }
    __syncthreads();
    const int bmu = s_bmu, sec = s_sec;

    // ---------- Phase 4: edge age/connect/prune (row+col bmu) + nb snapshot ----------
    if (tid < MAXN) {
      const int j = tid;
      int a = edges[(size_t)bmu * MAXN + j];
      s_nb[j] = (a > 0) ? 1 : 0;        // neighbor mask BEFORE connect/prune (ref step 3)
      if (a > 0) a += 1;                // step 3: age
      if (j == sec) a = 1;              // step 6: connect / reset age
      if (a > A_MAXE) a = 0;            // step 7: prune (only row/col bmu can exceed)
      edges[(size_t)bmu * MAXN + j] = a;
      edges[(size_t)j * MAXN + bmu] = a; // symmetric
    }
    // ---------- Phase 5: diff from PRE-update BMU position ----------
    if (tid < DIM) s_diff[tid] = s_img[tid] - nodes[(size_t)bmu * DIM + tid];
    __syncthreads();

    // ---------- Phase 6: move BMU (E_B) and neighbors (E_N); accumulate error ----------
    for (int tl = 0; tl < 16; ++tl) {
      const int n = wv * 16 + tl;
      const float coef = (n == bmu) ? E_B : (s_nb[n] ? E_N : 0.f);
      if (coef != 0.f) {
        float* p = nodes + (size_t)n * DIM;
        for (int i = lane; i < DIM; i += 32) p[i] += coef * s_diff[i];
      }
    }
    if (tid == 0) s_err[bmu] += s_d2[bmu];
    __syncthreads();

    // ---------- Phase 7: lonely detection (row sums) + park at 999 ----------
    {
      const int na = s_nact;
      for (int tl = 0; tl < 16; ++tl) {
        const int n = wv * 16 + tl;
        const int* er = edges + (size_t)n * MAXN;
        int s = 0;
        for (int i = lane; i < MAXN; i += 32) s += er[i];
        for (int off = 16; off; off >>= 1) s += __shfl_xor(s, off, 32);
        if (n < na && s == 0) {
          float* p = nodes + (size_t)n * DIM;
          for (int i = lane; i < DIM; i += 32) p[i] = 999.0f;
          if (lane == 0) s_err[n] = 0.f;
        }
      }
    }
    __syncthreads();

    // ---------- Phase 8: periodic insertion (cnt = t+1) ----------
    if (((t + 1) % L_INS) == 0 && s_nact < MAXN) {
      if (wv == 0) {
        const int na = s_nact;
        float bv = -INF; int bi = 0x7fffffff;
        for (int i = lane; i < MAXN; i += 32) {
          const float e = (i < na) ? s_err[i] : -INF;
          if (gt2(e, i, bv, bi)) { bv = e; bi = i; }
        }
        for (int off = 16; off; off >>= 1) {
          const float ov = __shfl_xor(bv, off, 32); const int oi = __shfl_xor(bi, off, 32);
          if (gt2(ov, oi, bv, bi)) { bv = ov; bi = oi; }
        }
        const int q = bi;
        float fv = -INF; int fi = 0x7fffffff;
        const int* eq = edges + (size_t)q * MAXN;
        for (int i = lane; i < MAXN; i += 32) {
          const float e = (eq[i] > 0) ? s_err[i] : -INF;
          if (gt2(e, i, fv, fi)) { fv = e; fi = i; }
        }
        for (int off = 16; off; off >>= 1) {
          const float ov = __shfl_xor(fv, off, 32); const int oi = __shfl_xor(fi, off, 32);
          if (gt2(ov, oi, fv, fi)) { fv = ov; fi = oi; }
        }
        if (lane == 0) { s_q = q; s_f = fi; }
      }
      __syncthreads();
      {
        const int q = s_q, f = s_f, na = s_nact;
        const float* pq = nodes + (size_t)q * DIM;
        const float* pf = nodes + (size_t)f * DIM;
        float* pn = nodes + (size_t)na * DIM;
        for (int i = tid; i < DIM; i += BS) pn[i] = 0.5f * (pq[i] + pf[i]);
        if (tid == 0) s_err[na] = s_err[q];
      }
      __syncthreads();
      if (tid == 0) s_nact = s_nact + 1;
    }
    __syncthreads();

    // ---------- Phase 9: decay errors ----------
    if (tid < MAXN) s_err[tid] *= D_FACT;
    __syncthreads();
  }
}

extern "C" void kernel_launch(void* const* d_in, const int* in_sizes, int n_in,
                              void* d_out, int out_size, void* d_ws, size_t ws_size,
                              hipStream_t stream) {
  const float* images     = (const float*)d_in[0]; // [256, 784]
  const float* nodes_init = (const float*)d_in[1]; // [512, 784]
  const float* err_init   = (const float*)d_in[2]; // [512]
  const int*   edges_init = (const int*)d_in[3];   // [512, 512]
  const int*   n_act_init = (const int*)d_in[4];   // [1]

  float* nodes = (float*)d_out;   // working buffer == output (nodes)
  int*   edges = (int*)d_ws;      // 1 MB edge matrix in workspace

  // Re-initialize mutable state every call (graph-capture safe, deterministic)
  hipMemcpyAsync(nodes, nodes_init, sizeof(float) * MAXN * DIM,
                 hipMemcpyDeviceToDevice, stream);
  hipMemcpyAsync(edges, edges_init, sizeof(int) * MAXN * MAXN,
                 hipMemcpyDeviceToDevice, stream);

  // Sequential scan -> one persistent workgroup (32 wave32s) with LDS barriers
  gng_persistent<<<dim3(1), dim3(BS), 0, stream>>>(nodes, edges, images,
                                                   err_init, n_act_init);
}